// DotInteraction_39891656245395
// MI455X (gfx1250) — compile-verified
//
#include <hip/hip_runtime.h>

typedef __attribute__((ext_vector_type(2))) float v2f;
typedef __attribute__((ext_vector_type(8))) float v8f;

#define B_TOT   16384
#define NEMB    26
#define DIMD    128
#define NFEAT   27
#define NPAD    32
#define LSTRIDE 132          // 128 + 4 pad floats: kills LDS bank conflicts, keeps 16B align
#define NPAIR   351          // 27*26/2

// Async global->LDS copy of 16 bytes per lane: no VGPR landing registers,
// tracked with ASYNCcnt. the LDS byte address is taken from vdst; on AMDGPU the generic
// pointer's low 32 bits equal the LDS byte offset (shared aperture base has
// zero low bits), so truncation gives the correct VDST value.
__device__ __forceinline__ void async_copy16(void* lds_ptr, const void* gaddr) {
    unsigned lds_off = (unsigned)(unsigned long long)lds_ptr;
    asm volatile("global_load_async_to_lds_b128 %0, %1, off"
                 :: "v"(lds_off), "v"(gaddr)
                 : "memory");
}

__device__ __forceinline__ void wait_async_zero() {
    asm volatile("s_wait_asynccnt 0x0" ::: "memory");
}

__global__ __launch_bounds__(64)
void dot_interact_wmma(const float* __restrict__ dense,
                       const float* __restrict__ embs,
                       float* __restrict__ out) {
    __shared__ float sm[2 * NPAD * LSTRIDE];   // 33792 B

    const int wave  = threadIdx.x >> 5;
    const int lane  = threadIdx.x & 31;
    const int batch = blockIdx.x * 2 + wave;

    float* T = sm + wave * NPAD * LSTRIDE;

    // ---- Stage T into LDS via async copies: row 0 = dense[b], rows 1..26 = embs[b].
    // One 16B async copy per lane per row (32 lanes * 16B = 512B = one 128-float row).
    {
        async_copy16(T + 0 * LSTRIDE + lane * 4,
                     dense + (size_t)batch * DIMD + lane * 4);

        const float* ebase = embs + (size_t)batch * NEMB * DIMD + lane * 4;
        #pragma unroll
        for (int r = 1; r < NFEAT; ++r)
            async_copy16(T + r * LSTRIDE + lane * 4, ebase + (r - 1) * DIMD);

        // Zero-fill padding rows 27..31 with regular LDS stores (DScnt, barrier-covered).
        const float4 z = {0.f, 0.f, 0.f, 0.f};
        #pragma unroll
        for (int r = NFEAT; r < NPAD; ++r)
            *(float4*)(T + r * LSTRIDE + lane * 4) = z;
    }
    wait_async_zero();      // ASYNCcnt == 0: async LDS writes visible
    __syncthreads();        // DScnt + workgroup barrier for the zero-fill stores

    // WMMA f32 16x16x4 operand addressing (wave32):
    //   A 16x4: lane L -> m = L%16, khalf = L/16; a[v] = A[m][2*khalf + v]
    //   B 4x16: lane L -> n = L%16, khalf = L/16; b[v] = B[2*khalf + v][n] = T[n][...]
    const int mrow  = lane & 15;
    const int khalf = lane >> 4;

    float* outb = out + (size_t)batch * NPAIR;

    // Upper-triangle tiles of the 32x32 (padded) Gram matrix: (0,0), (0,1), (1,1)
    #pragma unroll
    for (int t = 0; t < 3; ++t) {
        const int ti = (t == 2) ? 1 : 0;
        const int tj = (t == 0) ? 0 : 1;

        const float* arow = T + (ti * 16 + mrow) * LSTRIDE + 2 * khalf;
        const float* brow = T + (tj * 16 + mrow) * LSTRIDE + 2 * khalf;

        v8f c = {};
        #pragma unroll 8
        for (int k = 0; k < DIMD; k += 4) {
            v2f a = *(const v2f*)(arow + k);
            v2f b = (ti == tj) ? a : *(const v2f*)(brow + k);
            // D = A x B + C ; emits v_wmma_f32_16x16x4_f32
            c = __builtin_amdgcn_wmma_f32_16x16x4_f32(
                    /*neg_a=*/false, a, /*neg_b=*/false, b,
                    /*c_mod=*/(short)0, c, /*reuse_a=*/false, /*reuse_b=*/false);
        }

        // C/D layout: c[r] = Z[ti*16 + r + 8*khalf][tj*16 + mrow]
        const int n_g = tj * 16 + mrow;
        #pragma unroll
        for (int r = 0; r < 8; ++r) {
            const int m_g = ti * 16 + r + 8 * khalf;
            if (m_g < n_g && n_g < NFEAT) {
                // strict-upper row-major pair index for N=27
                const int pidx = m_g * (2 * NFEAT - m_g - 1) / 2 + (n_g - m_g - 1);
                outb[pidx] = c[r];
            }
        }
    }
}

extern "C" void kernel_launch(void* const* d_in, const int* in_sizes, int n_in,
                              void* d_out, int out_size, void* d_ws, size_t ws_size,
                              hipStream_t stream) {
    const float* dense = (const float*)d_in[0];   // [16384, 128]
    const float* embs  = (const float*)d_in[1];   // [16384, 26, 128]
    float*       out   = (float*)d_out;           // [16384, 351]

    dim3 grid(B_TOT / 2);   // 2 batch rows (waves) per 64-thread block
    dim3 block(64);
    hipLaunchKernelGGL(dot_interact_wmma, grid, block, 0, stream, dense, embs, out);
}